// MultiHeadGATEAULayer_14413910245568
// MI455X (gfx1250) — compile-verified
//
#include <hip/hip_runtime.h>
#include <hip/hip_bf16.h>

// ---------------------------------------------------------------------------
// MultiHeadGATEAULayer for MI455X (gfx1250, wave32, WMMA)
//
// Dense GEMMs (K=128) use V_WMMA_F32_16X16X4_F32. Weights are staged once per
// 256-thread block into LDS (stride 132 floats -> conflict-free ds_load_b64
// fragments, shared by 8 waves). Dual-weight kernels bind both GEMMs sharing
// an A matrix to ONE launch: pass 0 computes A@B0, pass 1 computes A@B1 with
// the wave's 16 A-rows (8KB) L0-resident, so A (incl. the 327MB edge matrix)
// streams from HBM once. Each pass keeps only 8 accumulator tiles (64 VGPRs)
// to stay below the spill threshold observed with 16 concurrent tiles.
// ---------------------------------------------------------------------------

typedef float v2f __attribute__((ext_vector_type(2)));
typedef float v8f __attribute__((ext_vector_type(8)));

#define N_NODES 50000
#define N_EDGES 640000
#define NEG_SLOPE 0.2f
#define BSTRIDE 132   // LDS row stride (floats): dword banks = 4*col+khalf

// ---- ordered-int mapping for float atomic max -----------------------------
__device__ __forceinline__ int f2ord(float f) {
    int i = __float_as_int(f);
    return (i >= 0) ? i : (i ^ 0x7fffffff);
}
__device__ __forceinline__ float ord2f(int i) {
    return __int_as_float((i >= 0) ? i : (i ^ 0x7fffffff));
}

// ---- stage a 128x128 weight into LDS as [col][k], stride BSTRIDE ----------
// non-trans: B stored [k][col]  -> Bs[col*BSTRIDE + k]
// trans:     B stored [col][k]  -> Bs[col*BSTRIDE + k]
// Both read patterns are fully coalesced on the global side.
template <bool TRANSB>
__device__ __forceinline__ void stage_B(const float* __restrict__ B,
                                        float* __restrict__ Bs,
                                        int tid, int nthreads) {
    for (int i = tid; i < 128 * 128; i += nthreads) {
        const int hi = i >> 7, lo = i & 127;
        if (TRANSB) Bs[hi * BSTRIDE + lo] = B[i];   // hi=col, lo=k
        else        Bs[lo * BSTRIDE + hi] = B[i];   // hi=k,  lo=col
    }
}

// ---- one 16x128 WMMA strip: acc over K=128 from LDS-staged weight ---------
// A frag (16x4 f32, 2 VGPR): lane L -> row L%16, K = (L/16)*2 + {0,1}.
// C/D tile: lane L, vgpr v -> M = v + 8*(L>>4), N = L&15.
__device__ __forceinline__ void wmma_strip(const float* __restrict__ Arow,
                                           const float* __restrict__ Bs,
                                           float* __restrict__ C,
                                           const float* __restrict__ bias,
                                           int m0, int lane) {
    const int row   = lane & 15;
    const int khalf = (lane >> 4) * 2;        // 0 or 2

    v8f acc[8];
#pragma unroll
    for (int n = 0; n < 8; ++n)
        acc[n] = (v8f){0.f, 0.f, 0.f, 0.f, 0.f, 0.f, 0.f, 0.f};

    for (int k = 0; k < 128; k += 4) {
        const v2f a = *(const v2f*)(Arow + k + khalf);   // global_load_b64
#pragma unroll
        for (int n = 0; n < 8; ++n) {
            const v2f b = *(const v2f*)(Bs + (n * 16 + row) * BSTRIDE + k + khalf);
            acc[n] = __builtin_amdgcn_wmma_f32_16x16x4_f32(
                false, a, false, b, (short)0, acc[n], false, false);
        }
    }

    const int mOff = (lane >> 4) * 8;
    const int nIdx = lane & 15;
#pragma unroll
    for (int n = 0; n < 8; ++n) {
        const int colg = n * 16 + nIdx;
        const float bv = bias ? bias[colg] : 0.f;
#pragma unroll
        for (int v = 0; v < 8; ++v)
            C[(size_t)(m0 + mOff + v) * 128 + colg] = acc[n][v] + bv;
    }
}

// ---------------------------------------------------------------------------
// Dual GEMM, two sequential passes: C0 = A@B0 then C1 = A@B1.
// 256 threads = 8 waves; each wave owns one 16-row strip of A, which is
// L0-resident for the second pass -> A streams from HBM once.
// ---------------------------------------------------------------------------
__global__ void gemm2_k128_wmma(const float* __restrict__ A,
                                const float* __restrict__ B0,
                                const float* __restrict__ B1,
                                float* __restrict__ C0,
                                float* __restrict__ C1, int M) {
    extern __shared__ float smem[];
    float* Bs0 = smem;
    float* Bs1 = smem + 128 * BSTRIDE;

    const int tid = threadIdx.x;
    stage_B<false>(B0, Bs0, tid, blockDim.x);
    stage_B<false>(B1, Bs1, tid, blockDim.x);
    __syncthreads();

    const int lane = tid & 31;
    const int m0 = (blockIdx.x * (blockDim.x >> 5) + (tid >> 5)) * 16;
    if (m0 >= M) return;                      // wave-uniform, after barrier

    const float* __restrict__ Arow = A + (size_t)(m0 + (lane & 15)) * 128;
    wmma_strip(Arow, Bs0, C0, nullptr, m0, lane);   // pass 0: A from HBM
    wmma_strip(Arow, Bs1, C1, nullptr, m0, lane);   // pass 1: A from L0
}

// ---------------------------------------------------------------------------
// Single GEMM with transposed weight ([out,in]) + bias: C = A @ B^T + bias
// ---------------------------------------------------------------------------
__global__ void gemm1_k128_wmma_tb(const float* __restrict__ A,
                                   const float* __restrict__ B,
                                   float* __restrict__ C,
                                   const float* __restrict__ bias, int M) {
    extern __shared__ float smem[];
    float* Bs = smem;

    const int tid = threadIdx.x;
    stage_B<true>(B, Bs, tid, blockDim.x);
    __syncthreads();

    const int lane = tid & 31;
    const int m0 = (blockIdx.x * (blockDim.x >> 5) + (tid >> 5)) * 16;
    if (m0 >= M) return;                      // wave-uniform, after barrier

    const float* __restrict__ Arow = A + (size_t)(m0 + (lane & 15)) * 128;
    wmma_strip(Arow, Bs, C, bias, m0, lane);
}

// ---------------------------------------------------------------------------
// init: zero agg / sums, set segment-max to ord(-1e9)
// ---------------------------------------------------------------------------
__global__ void init_kernel(float* __restrict__ agg, float* __restrict__ sums,
                            int* __restrict__ maxsc) {
    const int i = blockIdx.x * blockDim.x + threadIdx.x;
    if (i < N_NODES * 128) agg[i] = 0.f;
    if (i < N_NODES * 8) {
        sums[i]  = 0.f;
        maxsc[i] = f2ord(-1e9f);
    }
}

// ---------------------------------------------------------------------------
// Pass E1: new_edge_feature = h_u[tgt]+h_v[src]+h_e[map]+h_glob[tgt]
// -> output 2, per-head leaky-relu logits (all 128 threads compute partials,
// 16-lane shfl_xor tree reduce), atomic segment max.
// ---------------------------------------------------------------------------
__global__ void edge_feature_kernel(const float* __restrict__ h_u,
                                    const float* __restrict__ h_v,
                                    const float* __restrict__ h_e,
                                    const float* __restrict__ h_glob,
                                    const int* __restrict__ tgt,
                                    const int* __restrict__ src,
                                    const int* __restrict__ emap,
                                    const float* __restrict__ aw,
                                    const float* __restrict__ ab,
                                    float* __restrict__ nef_out,
                                    float* __restrict__ scores,
                                    int* __restrict__ maxsc) {
    const int e = blockIdx.x;
    const int t = threadIdx.x;
    __shared__ float s[128];

    const int tg = tgt[e], sr = src[e], em = emap[e];
    const float v = h_u[(size_t)tg * 128 + t] + h_v[(size_t)sr * 128 + t] +
                    h_e[(size_t)em * 128 + t] + h_glob[(size_t)tg * 128 + t];
    nef_out[(size_t)e * 128 + t] = v;
    s[t] = v;
    __syncthreads();

    const int h = t >> 4, l = t & 15;        // head, lane-in-head
    float p = 0.f;
#pragma unroll
    for (int j = 0; j < 8; ++j) {
        const int k = l + 16 * j;
        p += s[k] * aw[h * 128 + k];
    }
#pragma unroll
    for (int off = 8; off > 0; off >>= 1)
        p += __shfl_xor(p, off, 16);
    if (l == 0) {
        const float acc = p + ab[h];
        const float sc = (acc >= 0.f) ? acc : NEG_SLOPE * acc;
        scores[(size_t)e * 8 + h] = sc;
        atomicMax(&maxsc[(size_t)tg * 8 + h], f2ord(sc));
    }
}

// ---------------------------------------------------------------------------
// Pass E2: exp(score - max[tgt]) in place + segment sum via atomicAdd
// ---------------------------------------------------------------------------
__global__ void exp_kernel(float* __restrict__ scores,
                           const int* __restrict__ tgt,
                           const int* __restrict__ maxsc,
                           float* __restrict__ sums) {
    const int i = blockIdx.x * blockDim.x + threadIdx.x;
    if (i >= N_EDGES * 8) return;
    const int e = i >> 3, h = i & 7;
    const int tg = tgt[e];
    const float m  = ord2f(maxsc[(size_t)tg * 8 + h]);
    const float ex = expf(scores[i] - m);
    scores[i] = ex;
    atomicAdd(&sums[(size_t)tg * 8 + h], ex);
}

// ---------------------------------------------------------------------------
// Pass E3: alpha-weighted message scatter-add into agg[tgt]
// ---------------------------------------------------------------------------
__global__ void aggregate_kernel(const float* __restrict__ h_h,
                                 const float* __restrict__ h_g,
                                 const float* __restrict__ scores,
                                 const float* __restrict__ sums,
                                 const int* __restrict__ tgt,
                                 const int* __restrict__ src,
                                 const int* __restrict__ emap,
                                 float* __restrict__ agg) {
    const int e = blockIdx.x;
    const int t = threadIdx.x;            // feature index 0..127
    const int h = t >> 4;                 // head = feature/16
    const int tg = tgt[e], sr = src[e], em = emap[e];
    const float alpha =
        scores[(size_t)e * 8 + h] / (sums[(size_t)tg * 8 + h] + 1e-10f);
    const float val =
        (h_h[(size_t)sr * 128 + t] + h_g[(size_t)em * 128 + t]) * alpha;
    atomicAdd(&agg[(size_t)tg * 128 + t], val);
}

// ---------------------------------------------------------------------------
// new_h = h_0 + agg + proj_global
// ---------------------------------------------------------------------------
__global__ void combine_kernel(const float* __restrict__ h0,
                               const float* __restrict__ agg,
                               const float* __restrict__ pg,
                               float* __restrict__ new_h) {
    const int i = blockIdx.x * blockDim.x + threadIdx.x;
    if (i < N_NODES * 128) new_h[i] = h0[i] + agg[i] + pg[i];
}

// ---------------------------------------------------------------------------
extern "C" void kernel_launch(void* const* d_in, const int* in_sizes, int n_in,
                              void* d_out, int out_size, void* d_ws,
                              size_t ws_size, hipStream_t stream) {
    const int N = N_NODES, E = N_EDGES;

    const float* X    = (const float*)d_in[0];   // node features   [N,128]
    const float* EF   = (const float*)d_in[1];   // edge features   [E,128]
    const int*   eidx = (const int*)d_in[2];     // [2,E]
    const int*   emap = (const int*)d_in[3];     // [E]
    const float* G    = (const float*)d_in[4];   // global node     [N,128]
    const float* Wv   = (const float*)d_in[5];
    const float* Wu   = (const float*)d_in[6];
    const float* We   = (const float*)d_in[7];
    const float* Wge  = (const float*)d_in[8];
    const float* Wh   = (const float*)d_in[9];
    const float* Wg   = (const float*)d_in[10];
    const float* W0   = (const float*)d_in[11];
    const float* Wgn  = (const float*)d_in[12];
    const float* Wout = (const float*)d_in[13];  // [out,in] -> B^T path
    const float* bout = (const float*)d_in[14];
    const float* aw   = (const float*)d_in[15];  // [8,128]
    const float* ab   = (const float*)d_in[16];  // [8]

    const int* tgt = eidx;
    const int* src = eidx + E;

    // outputs: new_final [N,128] then new_edge_feature [E,128]
    float* out_final = (float*)d_out;
    float* out_nef   = (float*)d_out + (size_t)N * 128;

    // ---- workspace carve (floats) ----
    float* ws   = (float*)d_ws;
    float* h_v  = ws;                 ws += (size_t)N * 128;
    float* h_u  = ws;                 ws += (size_t)N * 128;
    float* h_gl = ws;                 ws += (size_t)N * 128;
    float* h_h  = ws;                 ws += (size_t)N * 128;
    float* h_0  = ws;                 ws += (size_t)N * 128;
    float* pglb = ws;                 ws += (size_t)N * 128;
    float* agg  = ws;                 ws += (size_t)N * 128;
    float* newh = ws;                 ws += (size_t)N * 128;
    float* h_e  = ws;                 ws += (size_t)E * 128;
    float* h_g  = ws;                 ws += (size_t)E * 128;
    float* scrs = ws;                 ws += (size_t)E * 8;
    float* sums = ws;                 ws += (size_t)N * 8;
    int*   maxs = (int*)ws;           ws += (size_t)N * 8;

    // ---- init (agg, sums, segment max) ----
    init_kernel<<<(N * 128 + 255) / 256, 256, 0, stream>>>(agg, sums, maxs);

    // ---- dense projections: dual GEMMs sharing A ----
    const int nodeBlks = (N / 16 + 7) / 8;    // 8 waves/block, 16 rows/wave
    const int edgeBlks = (E / 16 + 7) / 8;
    const size_t lds2  = (size_t)2 * 128 * BSTRIDE * sizeof(float);
    const size_t lds1  = (size_t)128 * BSTRIDE * sizeof(float);

    gemm2_k128_wmma<<<nodeBlks, 256, lds2, stream>>>(X,  Wv,  Wu,  h_v,  h_u,  N);
    gemm2_k128_wmma<<<nodeBlks, 256, lds2, stream>>>(X,  Wh,  W0,  h_h,  h_0,  N);
    gemm2_k128_wmma<<<nodeBlks, 256, lds2, stream>>>(G,  Wge, Wgn, h_gl, pglb, N);
    gemm2_k128_wmma<<<edgeBlks, 256, lds2, stream>>>(EF, We,  Wg,  h_e,  h_g,  E);

    // ---- edge passes ----
    edge_feature_kernel<<<E, 128, 0, stream>>>(h_u, h_v, h_e, h_gl, tgt, src,
                                               emap, aw, ab, out_nef, scrs, maxs);

    exp_kernel<<<(E * 8 + 255) / 256, 256, 0, stream>>>(scrs, tgt, maxs, sums);

    aggregate_kernel<<<E, 128, 0, stream>>>(h_h, h_g, scrs, sums, tgt, src,
                                            emap, agg);

    // ---- combine + final projection (B transposed, +bias) ----
    combine_kernel<<<(N * 128 + 255) / 256, 256, 0, stream>>>(h_0, agg, pglb, newh);

    gemm1_k128_wmma_tb<<<nodeBlks, 256, lds1, stream>>>(newh, Wout, out_final,
                                                        bout, N);
}